// GCN_custom_58643483460180
// MI455X (gfx1250) — compile-verified
//
#include <hip/hip_runtime.h>
#include <hip/hip_bf16.h>

#define NN      45360
#define EE      362880
#define HH      512
#define NLAYERS 12
#define OUTF    256

typedef float v2f __attribute__((ext_vector_type(2)));
typedef float v8f __attribute__((ext_vector_type(8)));

__device__ __forceinline__ float lrelu(float x) { return x > 0.0f ? x : 0.01f * x; }

// ---------------- K1: gather ocean nodes ----------------
__global__ void k_gather(const float* __restrict__ sst, const int* __restrict__ idx,
                         float* __restrict__ x0) {
    int i = blockIdx.x * blockDim.x + threadIdx.x;
    if (i < NN) x0[i] = sst[idx[i]];
}

// ---------------- K2: layer-1 rank-1 SpMM collapses to scalar scatter ----------------
__global__ void k_edge1(const int* __restrict__ rows, const int* __restrict__ cols,
                        const float* __restrict__ vals, const float* __restrict__ x0,
                        float* __restrict__ a1) {
    int e = blockIdx.x * blockDim.x + threadIdx.x;
    if (e < EE) atomicAdd(&a1[rows[e]], vals[e] * x0[cols[e]]);
}

// ---------------- K3: x1[n,h] = x0[n] + leaky(a1[n]*W1[h] + b1[h]) ----------------
__global__ void k_x1(const float* __restrict__ x0, const float* __restrict__ a1,
                     const float* __restrict__ W1, const float* __restrict__ b1,
                     float* __restrict__ x1) {
    size_t i = (size_t)blockIdx.x * blockDim.x + threadIdx.x;
    if (i >= (size_t)NN * HH) return;
    int n = (int)(i >> 9);
    int h = (int)(i & 511);
    x1[i] = x0[n] + lrelu(a1[n] * W1[h] + b1[h]);
}

// ---------------- K4: dense GEMM  S = X(45360 x 512) @ W(512 x 512), f32 WMMA ----------------
// Block = 256 threads (8 waves) -> 128x128 output tile; wave w owns rows [w*16, w*16+16)
// of the block tile and all 128 columns (8 x v8f accumulators).
// K chunked by 32. A(128x32) staged row-major (stride 34). B(32x128) staged K-PAIR
// INTERLEAVED: element (k,col) at (k>>1)*288 + col*2 + (k&1), so each lane's B operand
// {B[k][col], B[k+1][col]} is one contiguous 8B-aligned v2f -> single ds_load_b64,
// conflict-free (lanes 0-15 banks 0..31, lanes 16-31 banks 32..63 via the +288 offset).
// Per K-step all 8 B pairs are loaded into registers before the 8 WMMAs so the DS
// loads pipeline ahead of the matrix ops instead of stalling pairwise.
#define ASTRIDE 34
#define BSTRIDE 288   // floats per K-pair row of interleaved B (128 cols * 2 + 32 pad)

__global__ void __launch_bounds__(256) k_gemm(const float* __restrict__ X,
                                              const float* __restrict__ W,
                                              float* __restrict__ S) {
    __shared__ float Abuf[128 * ASTRIDE];    // 17408 B
    __shared__ float Bbuf[16 * BSTRIDE];     // 18432 B

    const int rb   = blockIdx.x >> 2;        // 355 row blocks of 128 rows
    const int cb   = blockIdx.x & 3;         // 4 col blocks of 128 cols
    const int t    = threadIdx.x;
    const int wave = t >> 5;
    const int lane = t & 31;
    const int m    = lane & 15;
    const int kh   = lane >> 4;

    const bool waveValid = (rb * 8 + wave) < (NN / 16);
    const int colBase = cb * 128;

    v8f acc[8];
#pragma unroll
    for (int i = 0; i < 8; ++i) acc[i] = (v8f){0.f,0.f,0.f,0.f,0.f,0.f,0.f,0.f};

    for (int kc = 0; kc < HH; kc += 32) {
        // ---- stage A: rows rb*128..+127 (clamped), k = kc..kc+31 ----
#pragma unroll
        for (int i = 0; i < 4; ++i) {
            int f  = t + i * 256;             // 0..1023 float4 slots
            int r  = f >> 3;                  // 0..127
            int k4 = (f & 7) << 2;            // 0,4,...,28
            int grow = rb * 128 + r;
            if (grow > NN - 1) grow = NN - 1;
            float4 v = *(const float4*)(X + (size_t)grow * HH + kc + k4);
            float* dst = &Abuf[r * ASTRIDE + k4];
            dst[0] = v.x; dst[1] = v.y; dst[2] = v.z; dst[3] = v.w;
        }
        // ---- stage B (K-pair interleaved): k rows kc..kc+31, cols colBase..+127 ----
#pragma unroll
        for (int i = 0; i < 4; ++i) {
            int f  = t + i * 256;
            int kr = f >> 5;                  // 0..31
            int c4 = (f & 31) << 2;           // 0..124
            float4 v = *(const float4*)(W + (size_t)(kc + kr) * HH + colBase + c4);
            float* dst = &Bbuf[(kr >> 1) * BSTRIDE + c4 * 2 + (kr & 1)];
            dst[0] = v.x; dst[2] = v.y; dst[4] = v.z; dst[6] = v.w;
        }
        __syncthreads();

        const float* __restrict__ arow = &Abuf[(wave * 16 + m) * ASTRIDE];
#pragma unroll
        for (int kk = 0; kk < 32; kk += 4) {
            v2f a;
            a.x = arow[kk + 2 * kh];
            a.y = arow[kk + 2 * kh + 1];
            // (kk + 2*kh) is even; its K-pair row is (kk+2*kh)>>1
            const v2f* __restrict__ bp =
                (const v2f*)&Bbuf[((kk + 2 * kh) >> 1) * BSTRIDE + m * 2];
            v2f bv[8];
#pragma unroll
            for (int ct = 0; ct < 8; ++ct) bv[ct] = bp[ct * 16];   // 4x ds_load_2addr_b64
#pragma unroll
            for (int ct = 0; ct < 8; ++ct)
                acc[ct] = __builtin_amdgcn_wmma_f32_16x16x4_f32(
                    /*neg_a=*/false, a, /*neg_b=*/false, bv[ct],
                    /*c_mod=*/(short)0, acc[ct], /*reuse_a=*/false, /*reuse_b=*/false);
        }
        __syncthreads();
    }

    if (waveValid) {
        const int row0 = rb * 128 + wave * 16 + kh * 8;
#pragma unroll
        for (int ct = 0; ct < 8; ++ct) {
            const int col = colBase + ct * 16 + m;
#pragma unroll
            for (int j = 0; j < 8; ++j)
                S[(size_t)(row0 + j) * HH + col] = acc[ct][j];
        }
    }
}

// ---------------- K5: layer-2 edge SpMM, thread per (edge, 4 columns) ----------------
__global__ void k_edge2(const int* __restrict__ rows, const int* __restrict__ cols,
                        const float* __restrict__ vals, const float* __restrict__ S,
                        float* __restrict__ Out) {
    const int HV = HH / 4;                    // 128
    long long idx = (long long)blockIdx.x * blockDim.x + threadIdx.x;
    if (idx >= (long long)EE * HV) return;
    int e  = (int)(idx / HV);
    int h4 = (int)(idx % HV) * 4;
    float v = vals[e];
    const float4 g = *(const float4*)(S + (size_t)cols[e] * HH + h4);
    float* dst = Out + (size_t)rows[e] * HH + h4;
    atomicAdd(dst + 0, v * g.x);
    atomicAdd(dst + 1, v * g.y);
    atomicAdd(dst + 2, v * g.z);
    atomicAdd(dst + 3, v * g.w);
}

// ---------------- K6: fused residual + leaky + column-sum pooling ----------------
#define ROWS_PER_CHUNK 90                     // 45360 / 90 = 504 chunks exactly
__global__ void k_pool(const float* __restrict__ x1, const float* __restrict__ o2,
                       const float* __restrict__ bc, float* __restrict__ pooled) {
    int t = blockIdx.x * blockDim.x + threadIdx.x;
    const int chunks = NN / ROWS_PER_CHUNK;
    if (t >= HH * chunks) return;
    int h     = t & 511;
    int chunk = t >> 9;
    float bch = bc[h];
    float s = 0.0f;
    int r0 = chunk * ROWS_PER_CHUNK;
    for (int r = r0; r < r0 + ROWS_PER_CHUNK; ++r) {
        size_t i = (size_t)r * HH + h;
        s += x1[i] + lrelu(o2[i] + bch);
    }
    atomicAdd(&pooled[h], s);
}

// ---------------- K7: gamma/beta heads (6144 dot products of length 512) ----------------
__global__ void k_heads(const float* __restrict__ pooled,
                        const float* __restrict__ Wg, const float* __restrict__ bg,
                        const float* __restrict__ Wb, const float* __restrict__ bb,
                        float* __restrict__ out) {
    int t = blockIdx.x * blockDim.x + threadIdx.x;
    const int total = 2 * NLAYERS * OUTF;
    if (t >= total) return;
    int which = t / (NLAYERS * OUTF);
    int r     = t % (NLAYERS * OUTF);         // l*OUT + o
    const float* W = which ? Wb : Wg;
    const float* b = which ? bb : bg;
    const float* w = W + (size_t)r * HH;
    float s = 0.0f;
#pragma unroll 4
    for (int f = 0; f < HH; ++f) s += pooled[f] * w[f];
    out[t] = s * (1.0f / NN) + b[r];
}

extern "C" void kernel_launch(void* const* d_in, const int* in_sizes, int n_in,
                              void* d_out, int out_size, void* d_ws, size_t ws_size,
                              hipStream_t stream) {
    const float* sst      = (const float*)d_in[0];
    const int*   node_idx = (const int*)  d_in[1];
    const int*   adj_rows = (const int*)  d_in[2];
    const int*   adj_cols = (const int*)  d_in[3];
    const float* adj_vals = (const float*)d_in[4];
    const float* W1       = (const float*)d_in[5];
    const float* b1       = (const float*)d_in[6];
    const float* Wc       = (const float*)d_in[7];
    const float* bc       = (const float*)d_in[8];
    const float* Wg       = (const float*)d_in[9];
    const float* bg       = (const float*)d_in[10];
    const float* Wb       = (const float*)d_in[11];
    const float* bb       = (const float*)d_in[12];
    float* out = (float*)d_out;

    float* ws = (float*)d_ws;
    float* x0 = ws;                               // NN
    float* a1 = x0 + NN;                          // NN
    float* x1 = a1 + NN;                          // NN*HH
    float* s2 = x1 + (size_t)NN * HH;             // NN*HH
    float* o2 = s2 + (size_t)NN * HH;             // NN*HH
    float* pooled = o2 + (size_t)NN * HH;         // HH

    hipMemsetAsync(a1, 0, NN * sizeof(float), stream);
    hipMemsetAsync(o2, 0, (size_t)NN * HH * sizeof(float), stream);
    hipMemsetAsync(pooled, 0, HH * sizeof(float), stream);

    k_gather<<<(NN + 255) / 256, 256, 0, stream>>>(sst, node_idx, x0);
    k_edge1 <<<(EE + 255) / 256, 256, 0, stream>>>(adj_rows, adj_cols, adj_vals, x0, a1);

    {
        size_t total = (size_t)NN * HH;
        k_x1<<<(unsigned)((total + 255) / 256), 256, 0, stream>>>(x0, a1, W1, b1, x1);
    }

    {
        int rowBlocks = (NN / 16 + 7) / 8;        // 355 blocks of 128 rows
        k_gemm<<<rowBlocks * 4, 256, 0, stream>>>(x1, Wc, s2);
    }

    {
        long long total = (long long)EE * (HH / 4);
        k_edge2<<<(unsigned)((total + 255) / 256), 256, 0, stream>>>(adj_rows, adj_cols,
                                                                     adj_vals, s2, o2);
    }

    {
        int total = HH * (NN / ROWS_PER_CHUNK);
        k_pool<<<(total + 255) / 256, 256, 0, stream>>>(x1, o2, bc, pooled);
    }

    k_heads<<<(2 * NLAYERS * OUTF + 255) / 256, 256, 0, stream>>>(pooled, Wg, bg, Wb, bb, out);
}